// Node_Attn_head_70987219468868
// MI455X (gfx1250) — compile-verified
//
#include <hip/hip_runtime.h>
#include <hip/hip_bf16.h>
#include <math.h>

// Problem constants (reference: B=1, N=8192, Cin=128, Cout=64)
#define CIN  128
#define COUT 64
#define RSEG 32   // row segments for the column-sum (softmax denominator) pass
#define NSEG 4    // m-segments for the main WMMA pass (parallelism over the reduction)

typedef __attribute__((ext_vector_type(2))) float v2f;
typedef __attribute__((ext_vector_type(8))) float v8f;

__device__ __forceinline__ float leaky01(float x) { return fmaxf(x, 0.01f * x); }

// ---------------------------------------------------------------------------
// Kernel 1: per-node features.
//   fts[n][o] = dot(x[n,:], w1[o,:])        (seq_fts, stored node-major)
//   f[n]      = dot(fts[n,:], w2)
// ---------------------------------------------------------------------------
__global__ void __launch_bounds__(COUT) feat_kernel(const float* __restrict__ x,
                                                    const float* __restrict__ w1,
                                                    const float* __restrict__ w2,
                                                    float* __restrict__ fts,
                                                    float* __restrict__ f) {
  __shared__ float xs[CIN];
  __shared__ float red[COUT];
  const int n = blockIdx.x;
  const int o = threadIdx.x;
  xs[o]        = x[(size_t)n * CIN + o];
  xs[o + COUT] = x[(size_t)n * CIN + o + COUT];
  __syncthreads();
  const float* __restrict__ wr = w1 + (size_t)o * CIN;
  float acc = 0.f;
#pragma unroll 8
  for (int i = 0; i < CIN; ++i) acc = fmaf(xs[i], wr[i], acc);
  fts[(size_t)n * COUT + o] = acc;
  red[o] = acc * w2[o];
  __syncthreads();
  for (int s = COUT >> 1; s > 0; s >>= 1) {
    if (o < s) red[o] += red[o + s];
    __syncthreads();
  }
  if (o == 0) f[n] = red[0];
}

// ---------------------------------------------------------------------------
// Kernel 2: softmax denominators (streaming pass 1 over bias, 268 MB).
//   psum[rs][m] = sum over rows r in segment rs of exp(leaky(f[r]+f[m]) + bias[r][m])
// Thread t handles one column; a block reads contiguous 1 KB rows -> coalesced.
// ---------------------------------------------------------------------------
__global__ void __launch_bounds__(256) colsum_kernel(const float* __restrict__ bias,
                                                     const float* __restrict__ f,
                                                     float* __restrict__ psum, int N) {
  const int cblocks = N / 256;
  const int m  = (blockIdx.x % cblocks) * 256 + threadIdx.x;
  const int rs = blockIdx.x / cblocks;
  const int rowsPer = N / RSEG;
  const float fm = f[m];
  float s = 0.f;
  const int r0 = rs * rowsPer;
  for (int r = r0; r < r0 + rowsPer; ++r) {
    float z = leaky01(f[r] + fm) + bias[(size_t)r * N + m];
    s += __expf(z);
  }
  psum[(size_t)rs * N + m] = s;
}

// ---------------------------------------------------------------------------
// Kernel 3: reduce partials and fold 1/S[m] into the B matrix:
//   ftsS[m][o] = fts[m][o] / S[m]
// Thread = (m,o) element -> coalesced read/write; the per-m psum sum is a
// 64-way broadcast (cache hit), total traffic is tiny (~5 MB).
// ---------------------------------------------------------------------------
__global__ void __launch_bounds__(256) scale_kernel(const float* __restrict__ psum,
                                                    const float* __restrict__ fts,
                                                    float* __restrict__ ftsS, int N) {
  const int idx = blockIdx.x * 256 + threadIdx.x;   // m*COUT + o
  const int m = idx >> 6;
  float s = 0.f;
#pragma unroll
  for (int rs = 0; rs < RSEG; ++rs) s += psum[(size_t)rs * N + m];
  ftsS[idx] = fts[idx] * (1.f / s);
}

// ---------------------------------------------------------------------------
// Kernel 4: main fused attention GEMM (streaming pass 2 over bias).
// One wave computes a 32(n) x 64(o) output tile for one m-segment:
//   ret[n,o] += sum_m exp(leaky(f[n]+f[m]) + bias[n,m]) * ftsS[m][o]
// Two stacked 16-row A tiles share the same four 4x16 B tiles, so each loop
// chunk streams 512 B of bias and issues 8x v_wmma_f32_16x16x4_f32.
// A layout (f32 16x4): lane<16 -> (M=lane, K=0/1) in v0/v1; lane>=16 -> K=2/3.
// C/D layout: VGPR r -> row r (+8 for upper half-wave), col = lane&15.
// ---------------------------------------------------------------------------
__global__ void __launch_bounds__(256) attn_wmma_kernel(const float* __restrict__ bias,
                                                        const float* __restrict__ f,
                                                        const float* __restrict__ ftsS,
                                                        float* __restrict__ partial, int N) {
  const int lane   = threadIdx.x & 31;
  const int wave   = (blockIdx.x << 3) | (threadIdx.x >> 5);
  const int nTiles = N >> 5;                 // 32-row tiles
  const int seg    = wave / nTiles;          // m-segment
  const int tile   = wave - seg * nTiles;
  const int n0     = tile << 5;
  const int row    = lane & 15;              // A: M index, B/D: N index
  const int kk     = (lane >> 4) << 1;       // K base: 0 for lanes 0-15, 2 for 16-31
  const float fn0  = f[n0 + row];
  const float fn1  = f[n0 + 16 + row];
  const int mChunk = N / NSEG;
  const int mBeg   = seg * mChunk;

  v8f aA0 = {}, aA1 = {}, aA2 = {}, aA3 = {};   // rows n0 .. n0+15
  v8f aB0 = {}, aB1 = {}, aB2 = {}, aB3 = {};   // rows n0+16 .. n0+31
  const float* __restrict__ biasRow0 = bias + (size_t)(n0 + row) * N;
  const float* __restrict__ biasRow1 = biasRow0 + (size_t)16 * N;
  const float* __restrict__ bBase    = ftsS + row;

  for (int m0 = mBeg; m0 < mBeg + mChunk; m0 += 4) {
    const int m = m0 + kk;                       // even -> 8B-aligned v2f loads
    const v2f bias0 = *reinterpret_cast<const v2f*>(biasRow0 + m);
    const v2f bias1 = *reinterpret_cast<const v2f*>(biasRow1 + m);
    const v2f fm    = *reinterpret_cast<const v2f*>(f + m);
    v2f a0, a1;
    a0.x = __expf(leaky01(fn0 + fm.x) + bias0.x);
    a0.y = __expf(leaky01(fn0 + fm.y) + bias0.y);
    a1.x = __expf(leaky01(fn1 + fm.x) + bias1.x);
    a1.y = __expf(leaky01(fn1 + fm.y) + bias1.y);
    const float* __restrict__ bp = bBase + (size_t)m * COUT;
    v2f bt0, bt1, bt2, bt3;
    bt0.x = bp[0];  bt0.y = bp[COUT + 0];
    bt1.x = bp[16]; bt1.y = bp[COUT + 16];
    bt2.x = bp[32]; bt2.y = bp[COUT + 32];
    bt3.x = bp[48]; bt3.y = bp[COUT + 48];
    aA0 = __builtin_amdgcn_wmma_f32_16x16x4_f32(false, a0, false, bt0, (short)0, aA0, false, false);
    aA1 = __builtin_amdgcn_wmma_f32_16x16x4_f32(false, a0, false, bt1, (short)0, aA1, false, false);
    aA2 = __builtin_amdgcn_wmma_f32_16x16x4_f32(false, a0, false, bt2, (short)0, aA2, false, false);
    aA3 = __builtin_amdgcn_wmma_f32_16x16x4_f32(false, a0, false, bt3, (short)0, aA3, false, false);
    aB0 = __builtin_amdgcn_wmma_f32_16x16x4_f32(false, a1, false, bt0, (short)0, aB0, false, false);
    aB1 = __builtin_amdgcn_wmma_f32_16x16x4_f32(false, a1, false, bt1, (short)0, aB1, false, false);
    aB2 = __builtin_amdgcn_wmma_f32_16x16x4_f32(false, a1, false, bt2, (short)0, aB2, false, false);
    aB3 = __builtin_amdgcn_wmma_f32_16x16x4_f32(false, a1, false, bt3, (short)0, aB3, false, false);
  }

  float* __restrict__ dst = partial + (size_t)seg * N * COUT;
  const int rhi = (lane >> 4) << 3;  // upper half-wave holds rows M=8..15
#pragma unroll
  for (int r = 0; r < 8; ++r) {
    const size_t b0 = (size_t)(n0 + rhi + r) * COUT + row;
    dst[b0 +  0] = aA0[r];
    dst[b0 + 16] = aA1[r];
    dst[b0 + 32] = aA2[r];
    dst[b0 + 48] = aA3[r];
    const size_t b1 = b0 + (size_t)16 * COUT;
    dst[b1 +  0] = aB0[r];
    dst[b1 + 16] = aB1[r];
    dst[b1 + 32] = aB2[r];
    dst[b1 + 48] = aB3[r];
  }
}

// ---------------------------------------------------------------------------
// Kernel 5: sum the NSEG partial results and apply the final LeakyReLU.
// ---------------------------------------------------------------------------
__global__ void __launch_bounds__(256) finalize_kernel(const float* __restrict__ partial,
                                                       float* __restrict__ out, int total) {
  const int i = blockIdx.x * 256 + threadIdx.x;
  if (i >= total) return;
  float s = 0.f;
#pragma unroll
  for (int g = 0; g < NSEG; ++g) s += partial[(size_t)g * total + i];
  out[i] = leaky01(s);
}

// ---------------------------------------------------------------------------
extern "C" void kernel_launch(void* const* d_in, const int* in_sizes, int n_in,
                              void* d_out, int out_size, void* d_ws, size_t ws_size,
                              hipStream_t stream) {
  (void)n_in; (void)out_size; (void)ws_size;
  const float* x    = (const float*)d_in[0];   // [1,N,128]
  const float* bias = (const float*)d_in[1];   // [1,N,N]
  const float* w1   = (const float*)d_in[2];   // [64,128]
  const float* w2   = (const float*)d_in[3];   // [64]
  float* out = (float*)d_out;                  // [1,N,64]
  const int N = in_sizes[0] / CIN;             // 8192

  // Workspace layout (floats): fts | f | psum | ftsS | partial  (~13.7 MB)
  float* ws   = (float*)d_ws;
  float* fts  = ws;                              // N*COUT
  float* f    = fts  + (size_t)N * COUT;         // N
  float* psum = f    + (size_t)N;                // RSEG*N
  float* ftsS = psum + (size_t)RSEG * N;         // N*COUT
  float* part = ftsS + (size_t)N * COUT;         // NSEG*N*COUT

  feat_kernel<<<N, COUT, 0, stream>>>(x, w1, w2, fts, f);
  colsum_kernel<<<(N / 256) * RSEG, 256, 0, stream>>>(bias, f, psum, N);
  scale_kernel<<<(N * COUT) / 256, 256, 0, stream>>>(psum, fts, ftsS, N);
  const int waves = NSEG * (N >> 5);             // 1024 waves
  attn_wmma_kernel<<<waves / 8, 256, 0, stream>>>(bias, f, ftsS, part, N);
  finalize_kernel<<<(N * COUT + 255) / 256, 256, 0, stream>>>(part, out, N * COUT);
}